// SlotAttention_50216757625627
// MI455X (gfx1250) — compile-verified
//
#include <hip/hip_runtime.h>
#include <math.h>

// ---------------------------------------------------------------------------
// SlotAttention on MI455X (gfx1250): fp32 WMMA (V_WMMA_F32_16X16X4_F32)
// B=32, N=4096, D=256, K=8 slots, ITERS=3, HID=256
// ---------------------------------------------------------------------------

typedef float v2f __attribute__((ext_vector_type(2)));
typedef float v8f __attribute__((ext_vector_type(8)));

#define D_DIM   256
#define N_DIM   4096
#define B_DIM   32
#define K_SLOT  8
#define SCALE_F 0.0625f   // 256^-0.5
#define EPS_F   1e-8f
#define LN_EPS_F 1e-5f

__device__ __forceinline__ v8f zero8() {
  v8f z = {0.f,0.f,0.f,0.f,0.f,0.f,0.f,0.f};
  return z;
}

// D = A(16x4) * B(4x16) + C, fp32 WMMA
__device__ __forceinline__ v8f wmma4(v2f a, v2f b, v8f c) {
  return __builtin_amdgcn_wmma_f32_16x16x4_f32(false, a, false, b, (short)0, c,
                                               false, false);
}

__device__ __forceinline__ float wave_sum(float v) {
  v += __shfl_xor(v, 16);
  v += __shfl_xor(v, 8);
  v += __shfl_xor(v, 4);
  v += __shfl_xor(v, 2);
  v += __shfl_xor(v, 1);
  return v;
}

// ---------------------------------------------------------------------------
// slots = mu + exp(logsigma) * noise       [B,K,D] (65536 threads)
// ---------------------------------------------------------------------------
__global__ void init_slots_kernel(const float* __restrict__ noise,
                                  const float* __restrict__ mu,
                                  const float* __restrict__ logsigma,
                                  float* __restrict__ slots) {
  int idx = blockIdx.x * blockDim.x + threadIdx.x;
  int d = idx & (D_DIM - 1);
  slots[idx] = mu[d] + __expf(logsigma[d]) * noise[idx];
}

__global__ void zero_kernel(float* __restrict__ p, int n) {
  int idx = blockIdx.x * blockDim.x + threadIdx.x;
  if (idx < n) p[idx] = 0.f;
}

// ---------------------------------------------------------------------------
// Generic LayerNorm over last dim (256): one wave per row, 8 rows per block.
// ---------------------------------------------------------------------------
__global__ void ln_rows_kernel(const float* __restrict__ in,
                               float* __restrict__ out,
                               const float* __restrict__ g,
                               const float* __restrict__ b) {
  int wave = threadIdx.x >> 5;
  int lane = threadIdx.x & 31;
  size_t row = (size_t)blockIdx.x * 8 + wave;
  const float* rp = in + row * D_DIM;
  float vals[8];
  float s = 0.f, sq = 0.f;
#pragma unroll
  for (int j = 0; j < 8; ++j) {
    int col = j * 32 + lane;
    float x = rp[col];
    vals[j] = x; s += x; sq += x * x;
  }
  s = wave_sum(s); sq = wave_sum(sq);
  float mean = s * (1.f / D_DIM);
  float var = sq * (1.f / D_DIM) - mean * mean;
  float inv = rsqrtf(var + LN_EPS_F);
  float* op = out + row * D_DIM;
#pragma unroll
  for (int j = 0; j < 8; ++j) {
    int col = j * 32 + lane;
    op[col] = (vals[j] - mean) * inv * g[col] + b[col];
  }
}

// ---------------------------------------------------------------------------
// Fused: x = LN(embeddings); keys = x@Wk^T; values = x@Wv^T
// block = 256 thr (8 waves) handles one batch's 32-row tile.
// Phase 1: LN 32x256 tile into LDS.  Phase 2: WMMA vs Wk and Wv.
// grid = B * N/32 = 4096
// ---------------------------------------------------------------------------
__global__ void ln_kv_kernel(const float* __restrict__ emb,
                             const float* __restrict__ g,
                             const float* __restrict__ b,
                             const float* __restrict__ Wk,
                             const float* __restrict__ Wv,
                             float* __restrict__ keys,
                             float* __restrict__ values) {
  __shared__ float xs[32 * D_DIM];
  int batch = blockIdx.x >> 7;            // N/32 = 128 tiles per batch
  int row0  = (blockIdx.x & 127) * 32;
  int wave = threadIdx.x >> 5;
  int lane = threadIdx.x & 31;

  // ---- phase 1: LayerNorm (wave per row, 4 rows per wave) ----
#pragma unroll
  for (int rr = 0; rr < 4; ++rr) {
    int rl = wave + rr * 8;
    const float* rp = emb + ((size_t)batch * N_DIM + row0 + rl) * D_DIM;
    float vals[8];
    float s = 0.f, sq = 0.f;
#pragma unroll
    for (int j = 0; j < 8; ++j) {
      int col = j * 32 + lane;
      float x = rp[col];
      vals[j] = x; s += x; sq += x * x;
    }
    s = wave_sum(s); sq = wave_sum(sq);
    float mean = s * (1.f / D_DIM);
    float var = sq * (1.f / D_DIM) - mean * mean;
    float inv = rsqrtf(var + LN_EPS_F);
#pragma unroll
    for (int j = 0; j < 8; ++j) {
      int col = j * 32 + lane;
      xs[rl * D_DIM + col] = (vals[j] - mean) * inv * g[col] + b[col];
    }
  }
  __syncthreads();

  // ---- phase 2: WMMA GEMM: wave -> 16 rows x 64 cols, K & V together ----
  int m_sub = wave >> 2;                  // 0..1
  int n_grp = wave & 3;                   // 0..3  (64 cols each)
  int lm = lane & 15;
  int kk = (lane >> 4) * 2;

  v8f accK[4], accV[4];
#pragma unroll
  for (int t = 0; t < 4; ++t) { accK[t] = zero8(); accV[t] = zero8(); }

  for (int k = 0; k < D_DIM; k += 4) {
    v2f a = *reinterpret_cast<const v2f*>(xs + (m_sub * 16 + lm) * D_DIM + k + kk);
#pragma unroll
    for (int t = 0; t < 4; ++t) {
      int col = n_grp * 64 + t * 16 + lm;
      v2f bk = *reinterpret_cast<const v2f*>(Wk + (size_t)col * D_DIM + k + kk);
      v2f bv = *reinterpret_cast<const v2f*>(Wv + (size_t)col * D_DIM + k + kk);
      accK[t] = wmma4(a, bk, accK[t]);
      accV[t] = wmma4(a, bv, accV[t]);
    }
  }

  int rhi = (lane >> 4) << 3;             // +8 rows for lanes 16..31
#pragma unroll
  for (int t = 0; t < 4; ++t) {
    int col = n_grp * 64 + t * 16 + lm;
#pragma unroll
    for (int r = 0; r < 8; ++r) {
      int row = row0 + m_sub * 16 + r + rhi;
      size_t o = ((size_t)batch * N_DIM + row) * D_DIM + col;
      keys[o]   = accK[t][r];
      values[o] = accV[t][r];
    }
  }
}

// ---------------------------------------------------------------------------
// Generic WMMA GEMM: out[M,Nout] = A[M,256] @ W[Nout,256]^T (+bias)(+epilogue)
// epi: 0 = none, 1 = leaky_relu(0.01), 2 = add residual `res`
// block = 256 thr; wave -> 16 rows x 64 cols (A fragment reused 4x)
// grid = (M/32) * (Nout/256)
// ---------------------------------------------------------------------------
__global__ void gemm_wmma_kernel(const float* __restrict__ A,
                                 const float* __restrict__ W,
                                 const float* __restrict__ bias,
                                 const float* __restrict__ res,
                                 float* __restrict__ out,
                                 int M, int Nout, int epi) {
  int nblocks = Nout >> 8;
  int mb = blockIdx.x / nblocks;
  int nb = blockIdx.x % nblocks;
  int wave = threadIdx.x >> 5;
  int lane = threadIdx.x & 31;
  int m_row = mb * 32 + (wave >> 2) * 16;
  int n_base = nb * 256 + (wave & 3) * 64;
  int lm = lane & 15;
  int kk = (lane >> 4) * 2;

  v8f acc[4];
#pragma unroll
  for (int t = 0; t < 4; ++t) acc[t] = zero8();

  const float* ap = A + (size_t)(m_row + lm) * D_DIM + kk;
  for (int k = 0; k < D_DIM; k += 4) {
    v2f a = *reinterpret_cast<const v2f*>(ap + k);
#pragma unroll
    for (int t = 0; t < 4; ++t) {
      int col = n_base + t * 16 + lm;
      v2f bf = *reinterpret_cast<const v2f*>(W + (size_t)col * D_DIM + k + kk);
      acc[t] = wmma4(a, bf, acc[t]);
    }
  }

  int rhi = (lane >> 4) << 3;
#pragma unroll
  for (int t = 0; t < 4; ++t) {
    int col = n_base + t * 16 + lm;
    float bv = bias ? bias[col] : 0.f;
#pragma unroll
    for (int r = 0; r < 8; ++r) {
      int row = m_row + r + rhi;
      float v = acc[t][r] + bv;
      if (epi == 1) v = (v > 0.f) ? v : 0.01f * v;
      else if (epi == 2) v += res[(size_t)row * Nout + col];
      out[(size_t)row * Nout + col] = v;
    }
  }
}

// ---------------------------------------------------------------------------
// dots = SCALE * q @ keys^T ; softmax over slot axis (in-register per lane),
// attn = softmax + EPS; accumulate rowsum over n (atomicAdd).
// q padded to 16 rows (rows 8..15 zero) in LDS.
// block = 256 thr (8 waves, 16 cols each = 128 cols); grid = B * 32
// ---------------------------------------------------------------------------
__global__ void dots_softmax_kernel(const float* __restrict__ queries,
                                    const float* __restrict__ keys,
                                    float* __restrict__ attn,
                                    float* __restrict__ rowsum,
                                    float* __restrict__ vis) {
  __shared__ float qs[16 * D_DIM];
  int batch = blockIdx.x >> 5;
  int nb    = blockIdx.x & 31;
  for (int i = threadIdx.x; i < 16 * D_DIM; i += 256) {
    int row = i >> 8;
    qs[i] = (row < K_SLOT) ? queries[(size_t)batch * K_SLOT * D_DIM + i] : 0.f;
  }
  __syncthreads();

  int wave = threadIdx.x >> 5;
  int lane = threadIdx.x & 31;
  int n0 = nb * 128 + wave * 16;
  int lm = lane & 15;
  int kk = (lane >> 4) * 2;

  v8f acc = zero8();
  for (int k = 0; k < D_DIM; k += 4) {
    v2f a = *reinterpret_cast<const v2f*>(qs + lm * D_DIM + k + kk);
    v2f bf = *reinterpret_cast<const v2f*>(
        keys + ((size_t)batch * N_DIM + n0 + lm) * D_DIM + k + kk);
    acc = wmma4(a, bf, acc);
  }

  // lane n (0..15) holds dots[b, 0..7, n0+n] in acc[0..7]
  float d[8], a8[8];
  float mx = -3.4e38f;
#pragma unroll
  for (int r = 0; r < 8; ++r) { d[r] = acc[r] * SCALE_F; mx = fmaxf(mx, d[r]); }
  float sum = 0.f;
#pragma unroll
  for (int r = 0; r < 8; ++r) { d[r] = __expf(d[r] - mx); sum += d[r]; }
  float inv = 1.f / sum;
#pragma unroll
  for (int r = 0; r < 8; ++r) a8[r] = d[r] * inv + EPS_F;

  if (lane < 16) {
#pragma unroll
    for (int r = 0; r < 8; ++r) {
      size_t o = ((size_t)batch * K_SLOT + r) * N_DIM + n0 + lane;
      attn[o] = a8[r];
      if (vis) vis[o] = a8[r];
    }
  }
#pragma unroll
  for (int r = 0; r < 8; ++r) {
    float v = (lane < 16) ? a8[r] : 0.f;
    v = wave_sum(v);
    if (lane == 0) atomicAdd(&rowsum[batch * K_SLOT + r], v);
  }
}

// ---------------------------------------------------------------------------
// updates[b,k,d] = (sum_n attn[b,k,n] * values[b,n,d]) / rowsum[b,k]
// A = attn (8 rows padded to 16), inner dim = N = 4096.
// block = 256 thr (8 waves x 16 cols = 128 cols); grid = B * 2
// ---------------------------------------------------------------------------
__global__ void updates_kernel(const float* __restrict__ attn,
                               const float* __restrict__ values,
                               const float* __restrict__ rowsum,
                               float* __restrict__ updates) {
  int batch = blockIdx.x >> 1;
  int half  = blockIdx.x & 1;
  int wave = threadIdx.x >> 5;
  int lane = threadIdx.x & 31;
  int col = half * 128 + wave * 16 + (lane & 15);
  int lm = lane & 15;
  int kk = (lane >> 4) * 2;

  v8f acc = zero8();
  for (int k = 0; k < N_DIM; k += 4) {
    v2f a;
    if (lm < K_SLOT)
      a = *reinterpret_cast<const v2f*>(
          attn + ((size_t)batch * K_SLOT + lm) * N_DIM + k + kk);
    else { a.x = 0.f; a.y = 0.f; }
    const float* vp = values + ((size_t)batch * N_DIM + k + kk) * D_DIM + col;
    v2f bf;
    bf.x = vp[0];
    bf.y = vp[D_DIM];
    acc = wmma4(a, bf, acc);
  }

  int rhi = (lane >> 4) << 3;
#pragma unroll
  for (int r = 0; r < 8; ++r) {
    int m = r + rhi;
    if (m < K_SLOT) {
      float rs = rowsum[batch * K_SLOT + m];
      updates[((size_t)batch * K_SLOT + m) * D_DIM + col] = acc[r] / rs;
    }
  }
}

// ---------------------------------------------------------------------------
// Pointwise GRU cell: slots = (1-z)*n + z*h   (reads gi[*,768], gh[*,768])
// ---------------------------------------------------------------------------
__global__ void gru_kernel(const float* __restrict__ gi,
                           const float* __restrict__ gh,
                           float* __restrict__ slots) {
  int idx = blockIdx.x * blockDim.x + threadIdx.x;   // 65536
  int row = idx >> 8;
  int col = idx & (D_DIM - 1);
  size_t base = (size_t)row * 768 + col;
  float i_r = gi[base], i_z = gi[base + 256], i_n = gi[base + 512];
  float h_r = gh[base], h_z = gh[base + 256], h_n = gh[base + 512];
  float r = 1.f / (1.f + __expf(-(i_r + h_r)));
  float z = 1.f / (1.f + __expf(-(i_z + h_z)));
  float n = tanhf(i_n + r * h_n);
  float h = slots[idx];
  slots[idx] = (1.f - z) * n + z * h;
}

// ---------------------------------------------------------------------------
// Host launch
// ---------------------------------------------------------------------------
extern "C" void kernel_launch(void* const* d_in, const int* in_sizes, int n_in,
                              void* d_out, int out_size, void* d_ws, size_t ws_size,
                              hipStream_t stream) {
  const float* emb       = (const float*)d_in[0];
  const float* noise     = (const float*)d_in[1];
  const float* mu        = (const float*)d_in[2];
  const float* logsigma  = (const float*)d_in[3];
  const float* Wq        = (const float*)d_in[4];
  const float* Wk        = (const float*)d_in[5];
  const float* Wv        = (const float*)d_in[6];
  const float* W_ih      = (const float*)d_in[7];
  const float* W_hh      = (const float*)d_in[8];
  const float* b_ih      = (const float*)d_in[9];
  const float* b_hh      = (const float*)d_in[10];
  const float* mlp_W1    = (const float*)d_in[11];
  const float* mlp_b1    = (const float*)d_in[12];
  const float* mlp_W2    = (const float*)d_in[13];
  const float* mlp_b2    = (const float*)d_in[14];
  const float* ln_in_g   = (const float*)d_in[15];
  const float* ln_in_b   = (const float*)d_in[16];
  const float* ln_sl_g   = (const float*)d_in[17];
  const float* ln_sl_b   = (const float*)d_in[18];
  const float* ln_ff_g   = (const float*)d_in[19];
  const float* ln_ff_b   = (const float*)d_in[20];

  const size_t NKV = (size_t)B_DIM * N_DIM * D_DIM;      // 33,554,432
  const size_t SL  = (size_t)B_DIM * K_SLOT * D_DIM;     // 65,536
  const size_t AT  = (size_t)B_DIM * K_SLOT * N_DIM;     // 1,048,576

  float* ws      = (float*)d_ws;
  float* keys    = ws;
  float* values  = keys + NKV;
  float* slots   = values + NKV;
  float* s_n     = slots + SL;
  float* queries = s_n + SL;
  float* attn    = queries + SL;
  float* rowsum  = attn + AT;
  float* upd     = rowsum + 256;
  float* gi      = upd + SL;
  float* gh      = gi + (size_t)256 * 768;
  float* y_ln    = gh + (size_t)256 * 768;
  float* hidden  = y_ln + SL;

  float* out_slots = (float*)d_out;
  float* out_attn  = out_slots + SL;

  // slot init
  init_slots_kernel<<<256, 256, 0, stream>>>(noise, mu, logsigma, slots);
  // fused LN(embeddings) + keys/values projections
  ln_kv_kernel<<<B_DIM * (N_DIM / 32), 256, 0, stream>>>(
      emb, ln_in_g, ln_in_b, Wk, Wv, keys, values);

  for (int it = 0; it < 3; ++it) {
    // queries = LN(slots) @ Wq^T
    ln_rows_kernel<<<32, 256, 0, stream>>>(slots, s_n, ln_sl_g, ln_sl_b);
    gemm_wmma_kernel<<<8, 256, 0, stream>>>(s_n, Wq, nullptr, nullptr,
                                            queries, 256, 256, 0);
    // dots + slot softmax + rowsums
    zero_kernel<<<1, 256, 0, stream>>>(rowsum, 256);
    dots_softmax_kernel<<<B_DIM * 32, 256, 0, stream>>>(
        queries, keys, attn, rowsum, (it == 2) ? out_attn : nullptr);
    // updates = normalize(attn) @ values
    updates_kernel<<<B_DIM * 2, 256, 0, stream>>>(attn, values, rowsum, upd);
    // GRU gates
    gemm_wmma_kernel<<<24, 256, 0, stream>>>(upd, W_ih, b_ih, nullptr,
                                             gi, 256, 768, 0);
    gemm_wmma_kernel<<<24, 256, 0, stream>>>(slots, W_hh, b_hh, nullptr,
                                             gh, 256, 768, 0);
    gru_kernel<<<256, 256, 0, stream>>>(gi, gh, slots);
    // residual MLP
    ln_rows_kernel<<<32, 256, 0, stream>>>(slots, y_ln, ln_ff_g, ln_ff_b);
    gemm_wmma_kernel<<<8, 256, 0, stream>>>(y_ln, mlp_W1, mlp_b1, nullptr,
                                            hidden, 256, 256, 1);
    float* dst = (it == 2) ? out_slots : slots;
    gemm_wmma_kernel<<<8, 256, 0, stream>>>(hidden, mlp_W2, mlp_b2, slots,
                                            dst, 256, 256, 2);
  }
}